// GATModel_17944373363172
// MI455X (gfx1250) — compile-verified
//
#include <hip/hip_runtime.h>
#include <hip/hip_bf16.h>
#include <math.h>

// ---------------------------------------------------------------------------
// GAT 2-layer forward for MI455X (gfx1250, wave32).
//   Layer1: h1 = x@W1 (N x 128), 2 heads x 64, edge softmax, aggregate, +b1, relu
//   Layer2: z  = h@W2 (N x 64), 1 head, edge softmax, aggregate, +b2
// Dense projections: LDS-staged fp32 WMMA (V_WMMA_F32_16X16X4_F32).
// Sparse phase: 3-pass segment softmax with L2-resident fp32 atomics
// (per-node tables are ~26 MB << 192 MB L2, so gathers/atomics stay on-chip).
// ---------------------------------------------------------------------------

#define NEG_SLOPE 0.2f

typedef float v2f __attribute__((ext_vector_type(2)));
typedef float v8f __attribute__((ext_vector_type(8)));

__device__ __forceinline__ float leaky(float x) {
    return x >= 0.f ? x : NEG_SLOPE * x;
}

// IEEE float atomic max: signed-int max for non-negative, unsigned-int min for
// negative values. Buffer must be initialized to -INFINITY (0xFF800000).
__device__ __forceinline__ void atomic_max_f32(float* addr, float val) {
    if (val >= 0.f) atomicMax((int*)addr, __float_as_int(val));
    else            atomicMin((unsigned int*)addr, __float_as_uint(val));
}

// ---------------------------------------------------------------------------
// fp32 WMMA GEMM with LDS staging: C[M,NC] = A[M,128] @ B[128,NC], row-major.
// grid.x = ceil(M/16); blockDim = 32*(NC/16)  (one wave per 16-col tile).
// A tile (16x128) staged once; B staged in two 64-row chunks (fits <64KB LDS).
// LDS padding tuned for wave32 banking:
//   As stride K+4  -> fragment loads hit banks {4r}/{4r+2}: 32 distinct banks
//   Bs stride NC+8 -> half-wave bank windows offset by 16: disjoint
// Fragment layout per ISA 7.12.2 (A 16x4 in 2 VGPRs, C/D 8 VGPRs).
// ---------------------------------------------------------------------------
template <int NC>
__global__ __launch_bounds__(32 * (NC / 16)) void wmma_gemm_lds(
    const float* __restrict__ A, const float* __restrict__ B,
    float* __restrict__ C, int M)
{
    constexpr int K   = 128;
    constexpr int KC  = 64;            // K-chunk staged in LDS at a time
    constexpr int LDA = K + 4;         // 132
    constexpr int LDB = NC + 8;
    constexpr int NT  = 32 * (NC / 16);

    __shared__ float As[16 * LDA];     // 8.25 KB
    __shared__ float Bs[KC * LDB];     // 34 KB (NC=128) / 18 KB (NC=64)

    const int tid = threadIdx.x;
    const int tm  = blockIdx.x;

    // ---- stage A tile (16 rows x 128), float4 granular, clamped rows ----
    for (int i = tid * 4; i < 16 * K; i += NT * 4) {
        const int rr = i / K, cc = i % K;
        int row = tm * 16 + rr; if (row >= M) row = M - 1;
        const float4 v = *(const float4*)(A + (size_t)row * K + cc);
        float* d = &As[rr * LDA + cc];
        d[0] = v.x; d[1] = v.y; d[2] = v.z; d[3] = v.w;
    }

    const int wave = tid >> 5;
    const int lane = tid & 31;
    const int r    = lane & 15;
    const int hw   = lane >> 4;         // half-wave 0/1
    const int col  = wave * 16 + r;

    v8f acc = {0.f, 0.f, 0.f, 0.f, 0.f, 0.f, 0.f, 0.f};

    for (int kc = 0; kc < K; kc += KC) {
        // ---- stage B chunk (KC rows x NC) ----
        const float* __restrict__ Bc = B + (size_t)kc * NC;
        for (int i = tid * 4; i < KC * NC; i += NT * 4) {
            const float4 v = *(const float4*)(Bc + i);
            const int rr = i / NC, cc = i % NC;
            float* d = &Bs[rr * LDB + cc];
            d[0] = v.x; d[1] = v.y; d[2] = v.z; d[3] = v.w;
        }
        __syncthreads();

#pragma unroll
        for (int kk = 0; kk < KC; kk += 4) {
            const int ka = kk + 2 * hw;
            v2f a, b;
            a.x = As[r * LDA + kc + ka];         // ds_load_b64 (8B aligned)
            a.y = As[r * LDA + kc + ka + 1];
            b.x = Bs[ka * LDB + col];
            b.y = Bs[(ka + 1) * LDB + col];
            acc = __builtin_amdgcn_wmma_f32_16x16x4_f32(
                /*neg_a=*/false, a, /*neg_b=*/false, b,
                /*c_mod=*/(short)0, acc, /*reuse_a=*/false, /*reuse_b=*/false);
        }
        __syncthreads();
    }

#pragma unroll
    for (int i = 0; i < 8; ++i) {
        const int ro = tm * 16 + hw * 8 + i;
        if (ro < M) C[(size_t)ro * NC + col] = acc[i];
    }
}

// ---------------------------------------------------------------------------
// Per-node attention coefficients: als[n,h] = <h[n,h,:], a_src[h,:]>, same dst.
// ---------------------------------------------------------------------------
__global__ void att_dots(const float* __restrict__ h,
                         const float* __restrict__ a_src,
                         const float* __restrict__ a_dst,
                         float* __restrict__ als, float* __restrict__ ald,
                         int n, int heads, int C)
{
    const int idx = blockIdx.x * blockDim.x + threadIdx.x;
    if (idx >= n * heads) return;
    const int node = idx / heads, hh = idx % heads;
    const float* __restrict__ hv = h + (size_t)node * heads * C + (size_t)hh * C;
    const float* __restrict__ as = a_src + (size_t)hh * C;
    const float* __restrict__ ad = a_dst + (size_t)hh * C;
    float s = 0.f, d = 0.f;
#pragma unroll 8
    for (int c = 0; c < C; ++c) { const float v = hv[c]; s += v * as[c]; d += v * ad[c]; }
    als[idx] = s;
    ald[idx] = d;
}

__global__ void fill_f32(float* __restrict__ p, long long n, float v)
{
    const long long i = (long long)blockIdx.x * blockDim.x + threadIdx.x;
    if (i < n) p[i] = v;
}

// ---------------------------------------------------------------------------
// Pass 1: segment max of leaky_relu(als[src] + ald[dst]) over dst (incl. loops)
// ---------------------------------------------------------------------------
__global__ void edge_max(const int* __restrict__ ei,
                         const float* __restrict__ als, const float* __restrict__ ald,
                         float* __restrict__ mx, int E, int n, int heads)
{
    const int idx = blockIdx.x * blockDim.x + threadIdx.x;
    const int tot = (E + n) * heads;
    if (idx >= tot) return;
    const int e = idx / heads, hh = idx % heads;
    int src, dst;
    if (e < E) { src = ei[e]; dst = ei[E + e]; } else { src = dst = e - E; }
    const float v = leaky(als[src * heads + hh] + ald[dst * heads + hh]);
    atomic_max_f32(&mx[dst * heads + hh], v);
}

// ---------------------------------------------------------------------------
// Pass 2: segment sum of exp(e - max[dst])
// ---------------------------------------------------------------------------
__global__ void edge_den(const int* __restrict__ ei,
                         const float* __restrict__ als, const float* __restrict__ ald,
                         const float* __restrict__ mx, float* __restrict__ den,
                         int E, int n, int heads)
{
    const int idx = blockIdx.x * blockDim.x + threadIdx.x;
    const int tot = (E + n) * heads;
    if (idx >= tot) return;
    const int e = idx / heads, hh = idx % heads;
    int src, dst;
    if (e < E) { src = ei[e]; dst = ei[E + e]; } else { src = dst = e - E; }
    const int ai = dst * heads + hh;
    const float v = leaky(als[src * heads + hh] + ald[ai]);
    atomicAdd(&den[ai], __expf(v - mx[ai]));
}

// ---------------------------------------------------------------------------
// Pass 3: agg[dst,:] += alpha * h[src,:].  256-thread blocks process 256/F
// edges (F = heads*C features per row); gathers and fp32 atomics stay
// row-contiguous -> coalesced, L2-resident.
// ---------------------------------------------------------------------------
template <int F>
__global__ __launch_bounds__(256) void edge_aggregate(
    const int* __restrict__ ei, const float* __restrict__ h,
    const float* __restrict__ als, const float* __restrict__ ald,
    const float* __restrict__ mx, const float* __restrict__ den,
    float* __restrict__ agg, int E, int n, int heads, int C)
{
    constexpr int EPB = 256 / F;
    const int e = blockIdx.x * EPB + threadIdx.x / F;   // edge id
    const int t = threadIdx.x % F;                      // feature id
    if (e >= E + n) return;
    const int hh = t / C;
    int src, dst;
    if (e < E) { src = ei[e]; dst = ei[E + e]; } else { src = dst = e - E; }
    const int ai = dst * heads + hh;
    const float v = leaky(als[src * heads + hh] + ald[ai]);
    const float alpha = __expf(v - mx[ai]) / den[ai];
    atomicAdd(&agg[(size_t)dst * F + t], alpha * h[(size_t)src * F + t]);
}

// out = relu(agg + b) (layer 1 epilogue)
__global__ void bias_relu(const float* __restrict__ agg, const float* __restrict__ b,
                          float* __restrict__ out, int n, int F)
{
    const long long i = (long long)blockIdx.x * blockDim.x + threadIdx.x;
    if (i >= (long long)n * F) return;
    const int c = (int)(i % F);
    const float v = agg[i] + b[c];
    out[i] = v > 0.f ? v : 0.f;
}

// out += b (layer 2 epilogue, in-place on d_out)
__global__ void bias_add(float* __restrict__ out, const float* __restrict__ b,
                         int n, int F)
{
    const long long i = (long long)blockIdx.x * blockDim.x + threadIdx.x;
    if (i >= (long long)n * F) return;
    out[i] = out[i] + b[(int)(i % F)];
}

// ---------------------------------------------------------------------------
extern "C" void kernel_launch(void* const* d_in, const int* in_sizes, int n_in,
                              void* d_out, int out_size, void* d_ws, size_t ws_size,
                              hipStream_t stream)
{
    const float* x   = (const float*)d_in[0];   // [N,128]
    const int*   ei  = (const int*)  d_in[1];   // [2,E]
    const float* W1  = (const float*)d_in[2];   // [128,128]
    const float* as1 = (const float*)d_in[3];   // [2,64]
    const float* ad1 = (const float*)d_in[4];   // [2,64]
    const float* b1  = (const float*)d_in[5];   // [128]
    const float* W2  = (const float*)d_in[6];   // [128,64]
    const float* as2 = (const float*)d_in[7];   // [1,64]
    const float* ad2 = (const float*)d_in[8];   // [1,64]
    const float* b2  = (const float*)d_in[9];   // [64]

    const int IN = 128, HID = 64, HEADS = 2, F1 = HEADS * HID /*128*/, OUT = 64;
    const int N    = in_sizes[0] / IN;
    const int E    = in_sizes[1] / 2;
    const int Etot = E + N;                     // edges + self loops

    float* out = (float*)d_out;                 // [N,64]

    // Workspace layout (floats). bufA/bufB reused across layers (~54 MB total).
    float* ws    = (float*)d_ws;
    float* bufA  = ws;                          // h1, then h2in (relu out) N*128
    float* bufB  = bufA + (size_t)N * F1;       // agg1, then z2           N*128
    float* als1v = bufB + (size_t)N * F1;       // N*2
    float* ald1v = als1v + (size_t)N * HEADS;
    float* mx1   = ald1v + (size_t)N * HEADS;
    float* den1  = mx1   + (size_t)N * HEADS;
    float* als2v = den1  + (size_t)N * HEADS;   // N
    float* ald2v = als2v + (size_t)N;
    float* mx2   = ald2v + (size_t)N;
    float* den2  = mx2   + (size_t)N;

    const int TB = 256;
    const int mtiles = (N + 15) / 16;

    // ---------------- Layer 1 ----------------
    wmma_gemm_lds<128><<<mtiles, 256, 0, stream>>>(x, W1, bufA, N);
    att_dots<<<(N * HEADS + TB - 1) / TB, TB, 0, stream>>>(bufA, as1, ad1, als1v, ald1v, N, HEADS, HID);

    fill_f32<<<(N * HEADS + TB - 1) / TB, TB, 0, stream>>>(mx1,  (long long)N * HEADS, -INFINITY);
    fill_f32<<<(N * HEADS + TB - 1) / TB, TB, 0, stream>>>(den1, (long long)N * HEADS, 0.f);
    fill_f32<<<((long long)N * F1 + TB - 1) / TB, TB, 0, stream>>>(bufB, (long long)N * F1, 0.f);

    edge_max<<<(Etot * HEADS + TB - 1) / TB, TB, 0, stream>>>(ei, als1v, ald1v, mx1, E, N, HEADS);
    edge_den<<<(Etot * HEADS + TB - 1) / TB, TB, 0, stream>>>(ei, als1v, ald1v, mx1, den1, E, N, HEADS);
    edge_aggregate<128><<<(Etot + 1) / 2, 256, 0, stream>>>(ei, bufA, als1v, ald1v, mx1, den1, bufB, E, N, HEADS, HID);

    // h2in = relu(agg1 + b1) -> overwrite bufA (h1 no longer needed)
    bias_relu<<<((long long)N * F1 + TB - 1) / TB, TB, 0, stream>>>(bufB, b1, bufA, N, F1);

    // ---------------- Layer 2 ----------------
    wmma_gemm_lds<64><<<mtiles, 128, 0, stream>>>(bufA, W2, bufB, N);
    att_dots<<<(N + TB - 1) / TB, TB, 0, stream>>>(bufB, as2, ad2, als2v, ald2v, N, 1, OUT);

    fill_f32<<<(N + TB - 1) / TB, TB, 0, stream>>>(mx2,  (long long)N, -INFINITY);
    fill_f32<<<(N + TB - 1) / TB, TB, 0, stream>>>(den2, (long long)N, 0.f);
    fill_f32<<<((long long)N * OUT + TB - 1) / TB, TB, 0, stream>>>(out, (long long)N * OUT, 0.f);

    edge_max<<<(Etot + TB - 1) / TB, TB, 0, stream>>>(ei, als2v, ald2v, mx2, E, N, 1);
    edge_den<<<(Etot + TB - 1) / TB, TB, 0, stream>>>(ei, als2v, ald2v, mx2, den2, E, N, 1);
    edge_aggregate<64><<<(Etot + 3) / 4, 256, 0, stream>>>(ei, bufB, als2v, ald2v, mx2, den2, out, E, N, 1, OUT);

    bias_add<<<((long long)N * OUT + TB - 1) / TB, TB, 0, stream>>>(out, b2, N, OUT);
}